// GCN_30451318129154
// MI455X (gfx1250) — compile-verified
//
#include <hip/hip_runtime.h>
#include <hip/hip_bf16.h>

typedef float v2f __attribute__((ext_vector_type(2)));
typedef float v8f __attribute__((ext_vector_type(8)));

// ---------------- zero ----------------
__global__ __launch_bounds__(256) void zero_kernel(float4* __restrict__ p, int n4) {
  int i = blockIdx.x * 256 + threadIdx.x;
  if (i < n4) p[i] = make_float4(0.f, 0.f, 0.f, 0.f);
}

// ---------------- degrees ----------------
__global__ __launch_bounds__(256) void degree_kernel(const int* __restrict__ src,
                                                     const int* __restrict__ dst,
                                                     unsigned* __restrict__ ds,
                                                     unsigned* __restrict__ dd, int nE) {
  int e = blockIdx.x * 256 + threadIdx.x;
  if (e < nE) {
    atomicAdd(&ds[src[e]], 1u);
    atomicAdd(&dd[dst[e]], 1u);
  }
}

// ---------------- norms: deg^-1/2 with clip(deg,1) ----------------
__global__ __launch_bounds__(256) void norm_kernel(const unsigned* __restrict__ ds,
                                                   const unsigned* __restrict__ dd,
                                                   float* __restrict__ sn,
                                                   float* __restrict__ dn, int n) {
  int i = blockIdx.x * 256 + threadIdx.x;
  if (i < n) {
    unsigned a = ds[i]; if (a < 1u) a = 1u;
    unsigned b = dd[i]; if (b < 1u) b = 1u;
    sn[i] = rsqrtf((float)a);
    dn[i] = rsqrtf((float)b);
  }
}

// ---------------- Y[n_rows x 64] = (X * row_scale[:,None]) @ W ----------------
// One wave computes a 16x64 output tile with V_WMMA_F32_16X16X4_F32.
// A 16x4 f32 layout: lanes 0-15 hold rows M=0..15; VGPR0 = K={0,2}, VGPR1 = K={1,3}
//   -> lane (L<16): (K0,K1), lane (L>=16): (K2,K3); contiguous pair => one b64 load.
// B 4x16 layout mirrors A (KxN): lane half selects K pair, lane&15 selects N.
// C 16x16 f32: VGPR j -> M=j (lanes 0-15) / M=j+8 (lanes 16-31), N = lane&15.
__global__ __launch_bounds__(256) void gemm64_wmma(const float* __restrict__ X,
                                                   const float* __restrict__ W,
                                                   const float* __restrict__ row_scale,
                                                   float* __restrict__ Y, int n_rows) {
  __shared__ float Wt[64 * 64];  // transposed: Wt[n*64 + k] = W[k*64 + n]
  const int tid = threadIdx.x;
  for (int i = tid; i < 64 * 64; i += 256) {
    int k = i >> 6, n = i & 63;
    Wt[n * 64 + k] = W[i];
  }
  __syncthreads();

  const int lane = tid & 31;
  const int wave = tid >> 5;
  const int tile = blockIdx.x * 8 + wave;  // 16-row tile index (wave-uniform branch)
  if (tile * 16 >= n_rows) return;

  const int m = lane & 15;
  const int khalf = lane >> 4;  // 0 or 1 -> K pair {0,1} or {2,3} within chunk
  const int row0 = tile * 16;
  const int row = row0 + m;
  const float sn = row_scale[row];
  const float* xrow = X + (size_t)row * 64;

  v8f acc0 = {}, acc1 = {}, acc2 = {}, acc3 = {};
#pragma unroll
  for (int kc = 0; kc < 16; ++kc) {
    const int kbase = kc * 4 + khalf * 2;
    const float2 av = *(const float2*)(xrow + kbase);
    v2f a; a.x = av.x * sn; a.y = av.y * sn;
    const v2f b0 = *(const v2f*)(&Wt[(m +  0) * 64 + kbase]);
    const v2f b1 = *(const v2f*)(&Wt[(m + 16) * 64 + kbase]);
    const v2f b2 = *(const v2f*)(&Wt[(m + 32) * 64 + kbase]);
    const v2f b3 = *(const v2f*)(&Wt[(m + 48) * 64 + kbase]);
    acc0 = __builtin_amdgcn_wmma_f32_16x16x4_f32(false, a, false, b0, (short)0, acc0, false, false);
    acc1 = __builtin_amdgcn_wmma_f32_16x16x4_f32(false, a, false, b1, (short)0, acc1, false, false);
    acc2 = __builtin_amdgcn_wmma_f32_16x16x4_f32(false, a, false, b2, (short)0, acc2, false, false);
    acc3 = __builtin_amdgcn_wmma_f32_16x16x4_f32(false, a, false, b3, (short)0, acc3, false, false);
  }

  float* yrow = Y + (size_t)row0 * 64;
  const int mofs = (lane < 16) ? 0 : 8;
#pragma unroll
  for (int j = 0; j < 8; ++j) {
    const size_t r = (size_t)(mofs + j) * 64;
    yrow[r +  0 + m] = acc0[j];
    yrow[r + 16 + m] = acc1[j];
    yrow[r + 32 + m] = acc2[j];
    yrow[r + 48 + m] = acc3[j];
  }
}

// ---------------- edge scatter-add: agg[dst] += hW[src] ----------------
// 32 threads per edge; each thread moves 2 consecutive floats (coalesced 256B/row).
__global__ __launch_bounds__(256) void scatter_kernel(const float* __restrict__ hW,
                                                      const int* __restrict__ src,
                                                      const int* __restrict__ dst,
                                                      float* __restrict__ agg, int nE) {
  int g = blockIdx.x * 256 + threadIdx.x;
  int e = g >> 5;
  if (e >= nE) return;
  int p = (g & 31) * 2;
  int s = src[e], d = dst[e];
  float2 v = *(const float2*)(hW + (size_t)s * 64 + p);
  float* o = agg + (size_t)d * 64 + p;
  atomicAdd(o, v.x);
  atomicAdd(o + 1, v.y);
}

// ---------------- epilogue: out = act(agg * dn[row] + b[col]); optionally zero agg ----------------
__global__ __launch_bounds__(256) void post_kernel(float* __restrict__ agg,
                                                   const float* __restrict__ dn,
                                                   const float* __restrict__ bias,
                                                   float* __restrict__ out,
                                                   int n_rows, int do_relu, int do_clear) {
  int g = blockIdx.x * 256 + threadIdx.x;  // n_rows*16 threads, float4 each
  if (g >= n_rows * 16) return;
  int row = g >> 4;
  int c = (g & 15) * 4;
  float4 v = *(const float4*)(agg + (size_t)row * 64 + c);
  float4 b = *(const float4*)(bias + c);
  float s = dn[row];
  float4 y;
  y.x = v.x * s + b.x; y.y = v.y * s + b.y;
  y.z = v.z * s + b.z; y.w = v.w * s + b.w;
  if (do_relu) {
    y.x = fmaxf(y.x, 0.f); y.y = fmaxf(y.y, 0.f);
    y.z = fmaxf(y.z, 0.f); y.w = fmaxf(y.w, 0.f);
  }
  *(float4*)(out + (size_t)row * 64 + c) = y;
  if (do_clear) *(float4*)(agg + (size_t)row * 64 + c) = make_float4(0.f, 0.f, 0.f, 0.f);
}

extern "C" void kernel_launch(void* const* d_in, const int* in_sizes, int n_in,
                              void* d_out, int out_size, void* d_ws, size_t ws_size,
                              hipStream_t stream) {
  const float* in_feat = (const float*)d_in[0];
  const int*   src     = (const int*)d_in[1];
  const int*   dst     = (const int*)d_in[2];
  const float* W1      = (const float*)d_in[3];
  const float* b1      = (const float*)d_in[4];
  const float* W2      = (const float*)d_in[5];
  const float* b2      = (const float*)d_in[6];
  float* out = (float*)d_out;

  const int N  = in_sizes[0] / 64;  // 100000 nodes
  const int nE = in_sizes[1];       // 1.6M edges

  const size_t feat_bytes = (size_t)N * 64 * sizeof(float);
  auto al = [](size_t x) { return (x + 255) & ~(size_t)255; };
  char* ws = (char*)d_ws;
  size_t off = 0;
  float*    hW    = (float*)(ws + off);    off += al(feat_bytes);
  float*    agg   = (float*)(ws + off);    off += al(feat_bytes);
  float*    h1    = (float*)(ws + off);    off += al(feat_bytes);
  unsigned* deg_s = (unsigned*)(ws + off); off += al((size_t)N * 4);
  unsigned* deg_d = (unsigned*)(ws + off); off += al((size_t)N * 4);
  float*    sn    = (float*)(ws + off);    off += al((size_t)N * 4);
  float*    dn    = (float*)(ws + off);    off += al((size_t)N * 4);
  (void)ws_size; (void)n_in; (void)out_size;

  // 1. zero agg + degree counters
  {
    int n4 = (int)(feat_bytes / 16);
    zero_kernel<<<(n4 + 255) / 256, 256, 0, stream>>>((float4*)agg, n4);
    int d4 = N / 4;
    zero_kernel<<<(d4 + 255) / 256, 256, 0, stream>>>((float4*)deg_s, d4);
    zero_kernel<<<(d4 + 255) / 256, 256, 0, stream>>>((float4*)deg_d, d4);
  }

  // 2. degrees, 3. norms
  degree_kernel<<<(nE + 255) / 256, 256, 0, stream>>>(src, dst, deg_s, deg_d, nE);
  norm_kernel<<<(N + 255) / 256, 256, 0, stream>>>(deg_s, deg_d, sn, dn, N);

  const int gemm_blocks = ((N + 15) / 16 + 7) / 8;
  const int scat_blocks = (nE * 32 + 255) / 256;
  const int post_blocks = (N * 16 + 255) / 256;

  // Layer 1: hW = (in_feat * sn) @ W1 ; agg = scatter ; h1 = relu(agg*dn + b1), clear agg
  gemm64_wmma<<<gemm_blocks, 256, 0, stream>>>(in_feat, W1, sn, hW, N);
  scatter_kernel<<<scat_blocks, 256, 0, stream>>>(hW, src, dst, agg, nE);
  post_kernel<<<post_blocks, 256, 0, stream>>>(agg, dn, b1, h1, N, /*relu=*/1, /*clear=*/1);

  // Layer 2: hW = (h1 * sn) @ W2 ; agg = scatter ; out = agg*dn + b2
  gemm64_wmma<<<gemm_blocks, 256, 0, stream>>>(h1, W2, sn, hW, N);
  scatter_kernel<<<scat_blocks, 256, 0, stream>>>(hW, src, dst, agg, nE);
  post_kernel<<<post_blocks, 256, 0, stream>>>(agg, dn, b2, out, N, /*relu=*/0, /*clear=*/0);
}